// WeightedHausdorffDistance_20907900796964
// MI455X (gfx1250) — compile-verified
//
#include <hip/hip_runtime.h>
#include <hip/hip_bf16.h>
#include <math.h>

// Problem constants (fixed by reference setup_inputs)
#define B_    4
#define H_    384
#define W_    384
#define M_    128
#define NPIX  (H_*W_)            // 147456 pixels per image
#define NTILES (NPIX/16)         // 9216 pixel tiles per image
#define TPW   32                 // pixel tiles processed per wave
#define WPI   (NTILES/TPW)       // 288 waves per image
#define TOTAL_WAVES (B_*WPI)     // 1152
#define EPS_   1e-6f
#define MAXD   543.0580080f      // sqrt(384^2 + 384^2)
#define FLTMAX 3.402823466e38f

typedef __attribute__((ext_vector_type(2))) float v2f;
typedef __attribute__((ext_vector_type(8))) float v8f;

__device__ __forceinline__ unsigned umin32(unsigned a, unsigned b) {
  return a < b ? a : b;   // -> v_min_u32 (valid float-min for non-negative floats)
}

// ---------------------------------------------------------------------------
// Init: per-GT global min buffer starts at max_dist (implements clip upper
// bound of term2). Per-wave partial buffers are fully overwritten, no init.
// ---------------------------------------------------------------------------
__global__ void whd_init_kernel(unsigned* __restrict__ gmin) {
  int i = blockIdx.x * blockDim.x + threadIdx.x;
  if (i < B_ * M_) gmin[i] = __float_as_uint(MAXD);
}

// ---------------------------------------------------------------------------
// Main: one wave32 owns TPW pixel tiles of one image. All 128 GT points live
// in registers as eight 16x4 f32 WMMA A-matrices:
//   A[m] = [-2*gy, -2*gx, gy^2+gx^2, 1]   (K=4)
//   B[n] = [  y  ,   x  ,    1     , y^2+x^2]
//   D[m][n] = A.B = ||loc_n - g_m||^2  exactly, in f32, C = 0.
// v_wmma_f32_16x16x4_f32 does the 1024-MAC d^2 tile; the nonlinearity is
// 1x v_med3_f32 + 1x v_sqrt_f32 (TRANS, co-executes) + 1x v_fma + 2x
// v_min_u32 per element (all mins on non-negative floats via uint ordering).
// ---------------------------------------------------------------------------
__global__ __launch_bounds__(256)
void whd_main_kernel(const float* __restrict__ prob,
                     const float* __restrict__ gt,
                     float* __restrict__ partS1,
                     float* __restrict__ partNP,
                     unsigned* __restrict__ gmin) {
  const int tid   = blockIdx.x * blockDim.x + threadIdx.x;
  const int wave  = tid >> 5;
  const int lane  = threadIdx.x & 31;
  const int b     = wave / WPI;
  const int tile0 = (wave % WPI) * TPW;
  const int nlo   = lane & 15;
  const bool hi   = lane >= 16;

  const float* g = gt   + b * (M_ * 2);   // [M][2] = (y, x)
  const float* p = prob + b * NPIX;

  // Resident A-matrices: 8 GT tiles x v2f = 16 VGPRs.
  // Lanes 0-15: VGPR0=K0, VGPR1=K1.  Lanes 16-31: VGPR0=K2, VGPR1=K3.
  v2f a[8];
#pragma unroll
  for (int t = 0; t < 8; ++t) {
    float gy = g[(t * 16 + nlo) * 2 + 0];
    float gx = g[(t * 16 + nlo) * 2 + 1];
    v2f av;
    av.x = hi ? (gy * gy + gx * gx) : (-2.0f * gy);
    av.y = hi ? 1.0f                : (-2.0f * gx);
    a[t] = av;
  }

  // Running column mins (term2) as float bit patterns: lane-half h covers GT
  // rows h*8..h*8+7 of every tile; each lane contributes its own pixel.
  unsigned minwu[64];
#pragma unroll
  for (int i = 0; i < 64; ++i) minwu[i] = __float_as_uint(MAXD);

  float s1 = 0.0f;   // sum p * min_j d
  float sp = 0.0f;   // sum p

  for (int tt = 0; tt < TPW; ++tt) {
    const int tile = tile0 + tt;
    const int pix  = tile * 16 + nlo;   // this lane's pixel (column n)
    const float fy = (float)(pix / W_);
    const float fx = (float)(pix % W_);

    // B-matrix (4x16): lanes 0-15 rows K0/K1, lanes 16-31 rows K2/K3.
    v2f bv;
    bv.x = hi ? 1.0f                : fy;
    bv.y = hi ? (fy * fy + fx * fx) : fx;

    const float pv = p[pix];
    if (tt + 1 < TPW) __builtin_prefetch(p + pix + 16, 0, 0); // next tile
    const float p2   = pv * pv;
    const float inv  = __builtin_amdgcn_rcpf(p2 * p2 + EPS_ / MAXD); // v_rcp_f32
    const float einv = EPS_ * inv;

    unsigned mind2u = __float_as_uint(FLTMAX);
    const v8f czero = {};
#pragma unroll
    for (int t = 0; t < 8; ++t) {
      // d2[m] = squared distance (pixel = this lane's column, GT row m)
      v8f d2 = __builtin_amdgcn_wmma_f32_16x16x4_f32(
          /*neg_a=*/false, a[t], /*neg_b=*/false, bv,
          /*c_mod=*/(short)0, czero, /*reuse_a=*/false, /*reuse_b=*/false);
#pragma unroll
      for (int m = 0; m < 8; ++m) {
        // clamp fp cancellation below 0 in a single v_med3_f32
        float dd = __builtin_amdgcn_fmed3f(d2[m], 0.0f, FLTMAX);
        unsigned du = __float_as_uint(dd);
        mind2u = umin32(mind2u, du);                       // row-min of d^2
        float dist = __builtin_amdgcn_sqrtf(dd);           // v_sqrt_f32 (TRANS)
        float w = fmaf(dist, inv, einv);  // (d + eps) / (p^4 + eps/maxd)
        minwu[t * 8 + m] = umin32(minwu[t * 8 + m], __float_as_uint(w));
      }
    }

    // term1: combine the two lane-halves (rows 0-7 vs 8-15 of each GT tile)
    unsigned md2u = umin32(mind2u, (unsigned)__shfl_xor((int)mind2u, 16, 32));
    float md = __builtin_amdgcn_sqrtf(__uint_as_float(md2u));
    s1 = fmaf(pv, md, s1);
    sp += pv;
  }

  // ---- term1 wave reduction (both halves computed identical values -> x0.5)
#pragma unroll
  for (int off = 16; off >= 1; off >>= 1) {
    s1 += __shfl_xor(s1, off, 32);
    sp += __shfl_xor(sp, off, 32);
  }
  if (lane == 0) {
    partS1[wave] = 0.5f * s1;   // deterministic per-wave slot
    partNP[wave] = 0.5f * sp;
  }

  // ---- term2: fold mins across the 16 pixels (lanes) of each half
#pragma unroll
  for (int i = 0; i < 64; ++i) {
#pragma unroll
    for (int off = 1; off <= 8; off <<= 1)
      minwu[i] = umin32(minwu[i], (unsigned)__shfl_xor((int)minwu[i], off, 32));
  }
  if (nlo == 0) {  // lane 0 -> GT rows 0-7, lane 16 -> GT rows 8-15
    const int halfOff = hi ? 8 : 0;
#pragma unroll
    for (int t = 0; t < 8; ++t)
#pragma unroll
      for (int m = 0; m < 8; ++m)
        atomicMin(&gmin[b * M_ + t * 16 + halfOff + m],
                  minwu[t * 8 + m]);  // w>=0: uint order == f32 order
  }
}

// ---------------------------------------------------------------------------
// Finalize: in-order sums of per-wave partials (deterministic), assemble the
// scalar. 2816 reads total — negligible.
// ---------------------------------------------------------------------------
__global__ void whd_final_kernel(const float* __restrict__ partS1,
                                 const float* __restrict__ partNP,
                                 const unsigned* __restrict__ gmin,
                                 float* __restrict__ out) {
  if (threadIdx.x == 0 && blockIdx.x == 0) {
    float acc = 0.0f;
    for (int b = 0; b < B_; ++b) {
      float s = 0.0f, n = 0.0f;
      for (int w = 0; w < WPI; ++w) {
        s += partS1[b * WPI + w];
        n += partNP[b * WPI + w];
      }
      float t1 = s / (n + EPS_);
      float tm = 0.0f;
      for (int j = 0; j < M_; ++j) tm += __uint_as_float(gmin[b * M_ + j]);
      acc += t1 + tm * (1.0f / (float)M_);
    }
    out[0] = acc * (1.0f / (float)B_);
  }
}

// ---------------------------------------------------------------------------
extern "C" void kernel_launch(void* const* d_in, const int* in_sizes, int n_in,
                              void* d_out, int out_size, void* d_ws, size_t ws_size,
                              hipStream_t stream) {
  const float* prob = (const float*)d_in[0];   // [B,H,W] f32
  const float* gt   = (const float*)d_in[1];   // [B,M,2] f32

  float*    partS1 = (float*)d_ws;                       // TOTAL_WAVES
  float*    partNP = partS1 + TOTAL_WAVES;               // TOTAL_WAVES
  unsigned* gmin   = (unsigned*)(partNP + TOTAL_WAVES);  // B_*M_

  whd_init_kernel<<<(B_ * M_ + 255) / 256, 256, 0, stream>>>(gmin);
  whd_main_kernel<<<TOTAL_WAVES / 8, 256, 0, stream>>>(prob, gt, partS1, partNP, gmin);
  whd_final_kernel<<<1, 32, 0, stream>>>(partS1, partNP, gmin, (float*)d_out);
}